// QuerySATConv_27144193311188
// MI455X (gfx1250) — compile-verified
//
#include <hip/hip_runtime.h>

// ----------------------------------------------------------------------------
// out[v,d] = deg[v] > 0 ? h[v,d] ** deg[v] : 0,  deg[v] = histogram(dst)
//
// Memory-bound: ~32 MB HBM traffic -> ~1.4 us at 23.3 TB/s, plus L2-atomic
// histogram. No matmul exists in this op, so no WMMA; we exercise the gfx1250
// async-to-LDS path (ASYNCcnt) for edge-index tiles, B128 vector memory ops
// for all streaming traffic, and native transcendentals (v_log_f32/v_exp_f32)
// for the integer-exponent pow.
// ----------------------------------------------------------------------------

#define D_FEAT 32
#define HIST_BLOCK 256            // 8 wave32 waves per block
#define EDGES_PER_THREAD 4        // one B128 op per lane
#define TILE_EDGES (HIST_BLOCK * EDGES_PER_THREAD)

// v4i32 pointers in explicit address spaces for the async-to-LDS builtin.
// (clang's gfx1250 builtin takes: global v4i32* src, LDS v4i32* dst, off, cpol)
typedef int int4v __attribute__((ext_vector_type(4)));
typedef __attribute__((address_space(1))) int4v g_int4v;   // global (AS1)
typedef __attribute__((address_space(3))) int4v l_int4v;   // LDS    (AS3)

// ---------------------------------------------------------------- zero deg --
__global__ void qsc_zero_deg(unsigned* __restrict__ deg, int n) {
    int i = blockIdx.x * blockDim.x + threadIdx.x;
    if (i < n) deg[i] = 0u;
}

// ------------------------------------------------------------ histogram ----
__global__ void qsc_hist(const int* __restrict__ dst,
                         unsigned* __restrict__ deg,
                         int n_edges) {
    __shared__ int tile[TILE_EDGES];
    const int t  = threadIdx.x;
    const int i0 = blockIdx.x * TILE_EDGES + t * EDGES_PER_THREAD;

#if defined(__gfx1250__) && __has_builtin(__builtin_amdgcn_global_load_async_to_lds_b128)
    // gfx1250 async copy: 16 B/lane global -> LDS, tracked by ASYNCcnt.
    if (i0 + EDGES_PER_THREAD <= n_edges) {
        __builtin_amdgcn_global_load_async_to_lds_b128(
            (g_int4v*)(dst + i0),
            (l_int4v*)(&tile[t * EDGES_PER_THREAD]),
            /*offset=*/0, /*cpol=*/0);
    } else {
        for (int k = 0; k < EDGES_PER_THREAD; ++k) {
            int i = i0 + k;
            tile[t * EDGES_PER_THREAD + k] = (i < n_edges) ? dst[i] : 0;
        }
    }
  #if __has_builtin(__builtin_amdgcn_s_wait_asynccnt)
    __builtin_amdgcn_s_wait_asynccnt(0);
  #else
    asm volatile("s_wait_asynccnt 0" ::: "memory");
  #endif
#else
    // Fallback: plain B128 load staged through LDS.
    if (i0 + EDGES_PER_THREAD <= n_edges) {
        const int4 v = *(const int4*)(dst + i0);
        tile[t * EDGES_PER_THREAD + 0] = v.x;
        tile[t * EDGES_PER_THREAD + 1] = v.y;
        tile[t * EDGES_PER_THREAD + 2] = v.z;
        tile[t * EDGES_PER_THREAD + 3] = v.w;
    } else {
        for (int k = 0; k < EDGES_PER_THREAD; ++k) {
            int i = i0 + k;
            tile[t * EDGES_PER_THREAD + k] = (i < n_edges) ? dst[i] : 0;
        }
    }
#endif
    __syncthreads();

    // Integer L2 atomics: commutative -> deterministic across graph replays.
    #pragma unroll
    for (int k = 0; k < EDGES_PER_THREAD; ++k) {
        int i = i0 + k;
        if (i < n_edges) {
            int v = tile[t * EDGES_PER_THREAD + k];
            atomicAdd(&deg[v], 1u);
        }
    }
}

// ---------------------------------------------------------------- pow map --
// One float4 (B128) per thread: thread t handles node v = t/8, feature chunk
// t%8. deg[] is L2-resident (400 KB), h/out are pure streams.
__device__ __forceinline__ float fast_exp2(float x) {
#if __has_builtin(__builtin_amdgcn_exp2f)
    return __builtin_amdgcn_exp2f(x);   // v_exp_f32
#else
    return exp2f(x);                    // device-lib exp2 == native v_exp_f32
#endif
}

__global__ void qsc_pow(const float* __restrict__ h,
                        const unsigned* __restrict__ deg,
                        float* __restrict__ out,
                        int n_vec4) {
    int t = blockIdx.x * blockDim.x + threadIdx.x;
    if (t >= n_vec4) return;
    const int v = t >> 3;                 // D_FEAT/4 == 8 chunks per node
    const unsigned d = deg[v];

    float4 o;
    if (d == 0u) {
        // DGL zero-fill for nodes with no in-edges.
        o.x = 0.f; o.y = 0.f; o.z = 0.f; o.w = 0.f;
    } else {
        const float4 hv = *(const float4*)(h + (size_t)t * 4);
        const float p = (float)d;
        // h in [0,1): h^p = exp2(p * log2(h)); log2(0) = -inf -> exp2 -> 0. OK.
        o.x = fast_exp2(p * __log2f(hv.x));
        o.y = fast_exp2(p * __log2f(hv.y));
        o.z = fast_exp2(p * __log2f(hv.z));
        o.w = fast_exp2(p * __log2f(hv.w));
    }
    *(float4*)(out + (size_t)t * 4) = o;
}

// ---------------------------------------------------------------- launch ---
extern "C" void kernel_launch(void* const* d_in, const int* in_sizes, int n_in,
                              void* d_out, int out_size, void* d_ws, size_t ws_size,
                              hipStream_t stream) {
    const float* h   = (const float*)d_in[0];   // [N, 32] f32
    const int*   dst = (const int*)d_in[1];     // [E] i32
    float*       out = (float*)d_out;           // [N, 32] f32

    const int n_nodes = in_sizes[0] / D_FEAT;
    const int n_edges = in_sizes[1];

    unsigned* deg = (unsigned*)d_ws;            // n_nodes u32 counters (400 KB)

    // 1) zero the degree counters (stateless across replays)
    qsc_zero_deg<<<(n_nodes + 255) / 256, 256, 0, stream>>>(deg, n_nodes);

    // 2) histogram of edge destinations via async-staged tiles + L2 atomics
    const int hist_blocks = (n_edges + TILE_EDGES - 1) / TILE_EDGES;
    qsc_hist<<<hist_blocks, HIST_BLOCK, 0, stream>>>(dst, deg, n_edges);

    // 3) elementwise pow, fully B128-vectorized
    const int n_vec4 = n_nodes * (D_FEAT / 4);
    qsc_pow<<<(n_vec4 + 255) / 256, 256, 0, stream>>>(h, deg, out, n_vec4);
}